// Top_k_Subs_55155970015737
// MI455X (gfx1250) — compile-verified
//
#include <hip/hip_runtime.h>
#include <math.h>

#define N_NODES   100000
#define IN_DIM    256
#define NUM_SUBS  20000
#define FLAT_N    640000
#define TOPK_N    10000

typedef __attribute__((ext_vector_type(2))) float v2f;
typedef __attribute__((ext_vector_type(8))) float v8f;

// ---------------------------------------------------------------------------
// Kernel 1: ragged gather + segment-sum. segment_ids are sorted, so each
// block (one per segment) binary-searches its [start,end) row range, then
// thread c accumulates column c across the range (coalesced across the block;
// x is 100MB and L2-resident on MI455X's 192MB L2, so the random row gather
// is L2-bandwidth bound, not HBM bound).
// ---------------------------------------------------------------------------
__global__ void seg_sum_kernel(const float* __restrict__ x,
                               const int*   __restrict__ sub_indices,
                               const int*   __restrict__ seg_ids,
                               float*       __restrict__ reps) {
    __shared__ int bounds[2];
    const int s = blockIdx.x;
    const int c = threadIdx.x;
    if (c == 0) {
        int lo = 0, hi = FLAT_N;
        while (lo < hi) { int mid = (lo + hi) >> 1; if (seg_ids[mid] < s) lo = mid + 1; else hi = mid; }
        bounds[0] = lo;
        hi = FLAT_N;
        while (lo < hi) { int mid = (lo + hi) >> 1; if (seg_ids[mid] < s + 1) lo = mid + 1; else hi = mid; }
        bounds[1] = lo;
    }
    __syncthreads();
    const int start = bounds[0], end = bounds[1];
    float acc = 0.0f;
    for (int r = start; r < end; ++r) {
        const int row = sub_indices[r];
        acc += x[(size_t)row * IN_DIM + c];
    }
    reps[(size_t)s * IN_DIM + c] = acc;   // empty segments get 0 (deterministic)
}

// ---------------------------------------------------------------------------
// Kernel 2: weights = reps @ w + b ; scores = sigmoid(weights).
// One wave32 per 16-row tile of reps. Uses V_WMMA_F32_16X16X4_F32 (full fp32
// precision, so the top-k ordering matches the f32 reference bit-for-bit).
// B is proj_w replicated across all 16 N columns -> every column of D holds
// the 16 dot products. ISA layouts (CDNA5 ISA):
//   A 16x4 f32 : lane L -> row L%16, K = (L>>4)*2 + vgpr
//   B  4x16 f32: lane L -> K = (L>>4)*2 + vgpr (N = L%16, replicated here)
//   D 16x16 f32: lane 0 col N=0 rows 0..7 in d[0..7]; lane 16 rows 8..15.
// blockDim == 32 and no divergence before the WMMA -> EXEC all ones.
// ---------------------------------------------------------------------------
__global__ void proj_sigmoid_kernel(const float* __restrict__ reps,
                                    const float* __restrict__ w,
                                    const float* __restrict__ bptr,
                                    float*       __restrict__ scores) {
    const int tile = blockIdx.x;                 // 20000/16 = 1250 tiles, exact
    const int lane = threadIdx.x;                // wave32
    const int m    = tile * 16 + (lane & 15);
    const int kk   = (lane >> 4) * 2;            // 0 for lanes 0-15, 2 for 16-31
    const float* rowp = reps + (size_t)m * IN_DIM;

    v8f d = {};
#pragma unroll 4
    for (int k0 = 0; k0 < IN_DIM; k0 += 4) {
        v2f a, b;
        a.x = rowp[k0 + kk];
        a.y = rowp[k0 + kk + 1];
        b.x = w[k0 + kk];
        b.y = w[k0 + kk + 1];
        d = __builtin_amdgcn_wmma_f32_16x16x4_f32(
                /*neg_a=*/false, a, /*neg_b=*/false, b,
                /*c_mod=*/(short)0, d, /*reuse_a=*/false, /*reuse_b=*/false);
    }

    const float bias = *bptr;
    if (lane == 0 || lane == 16) {               // column N=0 of D
        const int base = tile * 16 + (lane >> 4) * 8;
#pragma unroll
        for (int r = 0; r < 8; ++r) {
            const float wsum = d[r] + bias;
            scores[base + r] = 1.0f / (1.0f + expf(-wsum));
        }
    }
}

// ---------------------------------------------------------------------------
// Kernel 3: exact rank of each score (descending, stable tie-break by lower
// index first) == jax.lax.top_k ordering. O(N^2)=4e8 compares against
// LDS-tiled score chunks.
// ---------------------------------------------------------------------------
#define RANK_TILE 2048
__global__ void rank_kernel(const float* __restrict__ scores,
                            int*         __restrict__ rank) {
    __shared__ float sm[RANK_TILE];
    const int s = blockIdx.x * blockDim.x + threadIdx.x;
    const float my = (s < NUM_SUBS) ? scores[s] : 0.0f;
    int cnt = 0;
    for (int t0 = 0; t0 < NUM_SUBS; t0 += RANK_TILE) {
        const int n = (NUM_SUBS - t0 < RANK_TILE) ? (NUM_SUBS - t0) : RANK_TILE;
        __syncthreads();
        for (int j = threadIdx.x; j < n; j += blockDim.x) sm[j] = scores[t0 + j];
        __syncthreads();
        if (s < NUM_SUBS) {
            for (int j = 0; j < n; ++j) {
                const float v = sm[j];
                const int   t = t0 + j;
                cnt += (v > my) || (v == my && t < s);
            }
        }
    }
    if (s < NUM_SUBS) rank[s] = cnt;
}

// ---------------------------------------------------------------------------
// Kernel 4: scatter outputs. d_out = [ idx(float) x TOPK | reps[idx] x TOPK*256 ].
// ---------------------------------------------------------------------------
__global__ void emit_kernel(const float* __restrict__ reps,
                            const int*   __restrict__ rank,
                            float*       __restrict__ out_idx,
                            float*       __restrict__ out_sel) {
    const int s = blockIdx.x;
    const int r = rank[s];
    if (r < TOPK_N) {
        const int c = threadIdx.x;
        out_sel[(size_t)r * IN_DIM + c] = reps[(size_t)s * IN_DIM + c];
        if (c == 0) out_idx[r] = (float)s;
    }
}

extern "C" void kernel_launch(void* const* d_in, const int* in_sizes, int n_in,
                              void* d_out, int out_size, void* d_ws, size_t ws_size,
                              hipStream_t stream) {
    (void)in_sizes; (void)n_in; (void)out_size; (void)ws_size;

    const float* x           = (const float*)d_in[0];   // [100000, 256]
    const int*   sub_indices = (const int*)  d_in[1];   // [640000]
    const int*   seg_ids     = (const int*)  d_in[2];   // [640000] sorted
    const float* proj_w      = (const float*)d_in[3];   // [256]
    const float* proj_b      = (const float*)d_in[4];   // scalar

    // workspace layout: reps [20000*256 f32] | scores [20000 f32] | rank [20000 i32]
    float* reps   = (float*)d_ws;
    float* scores = reps + (size_t)NUM_SUBS * IN_DIM;
    int*   rank   = (int*)(scores + NUM_SUBS);

    float* out_idx = (float*)d_out;        // TOPK entries
    float* out_sel = out_idx + TOPK_N;     // TOPK * 256 entries

    seg_sum_kernel<<<NUM_SUBS, IN_DIM, 0, stream>>>(x, sub_indices, seg_ids, reps);
    proj_sigmoid_kernel<<<NUM_SUBS / 16, 32, 0, stream>>>(reps, proj_w, proj_b, scores);
    rank_kernel<<<(NUM_SUBS + 255) / 256, 256, 0, stream>>>(scores, rank);
    emit_kernel<<<NUM_SUBS, IN_DIM, 0, stream>>>(reps, rank, out_idx, out_sel);
}